// GCNLayer_68195490726427
// MI455X (gfx1250) — compile-verified
//
#include <hip/hip_runtime.h>

typedef __attribute__((ext_vector_type(2))) float v2f;
typedef __attribute__((ext_vector_type(8))) float v8f;

#define FEAT 128

// ---------------- degree count ----------------
__global__ void k_zero_u32(unsigned* __restrict__ p, int n) {
    int i = blockIdx.x * blockDim.x + threadIdx.x;
    if (i < n) p[i] = 0u;
}

__global__ void k_count_deg(const int* __restrict__ col, unsigned* __restrict__ cnt, int E) {
    int i = blockIdx.x * blockDim.x + threadIdx.x;
    if (i < E) atomicAdd(&cnt[col[i]], 1u);
}

__global__ void k_dinv(const unsigned* __restrict__ cnt, float* __restrict__ dinv, int N) {
    int i = blockIdx.x * blockDim.x + threadIdx.x;
    if (i < N) {
        float deg = 2.0f + (float)cnt[i];   // improved self-loop fill = 2.0, always > 0
        dinv[i] = rsqrtf(deg);
    }
}

// ---------------- xW = x @ W  (fp32 WMMA 16x16x4) ----------------
// One wave -> one 16x16 tile of xW. Block = 256 thr = 8 waves covering all
// 128 output columns for one 16-row stripe.
__global__ void __launch_bounds__(256)
k_gemm_xw(const float* __restrict__ x, const float* __restrict__ W,
          float* __restrict__ xW, int N) {
    const int wave  = threadIdx.x >> 5;
    const int lane  = threadIdx.x & 31;
    const int l15   = lane & 15;
    const int khalf = (lane >> 4) << 1;      // 0 for lanes 0-15, 2 for lanes 16-31
    const int row0  = blockIdx.x << 4;
    const int col0  = wave << 4;

    int arow = row0 + l15;
    if (arow >= N) arow = N - 1;             // clamp (no EXEC divergence before WMMA)
    const float* aptr = x + (size_t)arow * FEAT + khalf;
    const float* bptr = W + (size_t)khalf * FEAT + col0 + l15;

    v8f acc = {};
#pragma unroll
    for (int kk = 0; kk < FEAT; kk += 4) {
        // A 16x4 layout: lane l<16: v0=A[l][kk+0], v1=A[l][kk+1]; lane 16+l: K+2/K+3
        v2f a;
        a.x = aptr[kk];
        a.y = aptr[kk + 1];
        // B 4x16 layout: lane l<16: v0=B[kk+0][c0+l], v1=B[kk+1][c0+l]; lanes 16-31: K+2/K+3
        v2f b;
        b.x = bptr[(size_t)kk * FEAT];
        b.y = bptr[(size_t)(kk + 1) * FEAT];
        acc = __builtin_amdgcn_wmma_f32_16x16x4_f32(false, a, false, b,
                                                    (short)0, acc, false, false);
    }

    // D layout: VGPR j, lanes 0-15 -> M=j, lanes 16-31 -> M=j+8; N = lane&15
    const int mbase = row0 + ((lane >> 4) << 3);
    float* optr = xW + (size_t)mbase * FEAT + col0 + l15;
#pragma unroll
    for (int j = 0; j < 8; ++j) {
        if (mbase + j < N) optr[(size_t)j * FEAT] = acc[j];
    }
}

// ---------------- out = b + 2*dinv^2 * xW  (self-loop term) ----------------
__global__ void k_init_out(const float* __restrict__ xW, const float* __restrict__ dinv,
                           const float* __restrict__ bias, float* __restrict__ out, int N) {
    int i = blockIdx.x * blockDim.x + threadIdx.x;
    if (i < N * FEAT) {
        int n = i >> 7;
        int f = i & (FEAT - 1);
        float di = dinv[n];
        out[i] = bias[f] + 2.0f * di * di * xW[i];
    }
}

// ---------------- edge scatter: out[col] += dinv[row]*dinv[col]*xW[row] ----------------
__global__ void __launch_bounds__(256)
k_aggregate(const int* __restrict__ row, const int* __restrict__ col,
            const float* __restrict__ dinv, const float* __restrict__ xW,
            float* __restrict__ out, int E) {
    int gw   = (blockIdx.x * blockDim.x + threadIdx.x) >> 5;   // one wave per edge
    int lane = threadIdx.x & 31;
    if (gw >= E) return;
    int r = row[gw];
    int c = col[gw];
    float nrm = dinv[r] * dinv[c];

    const float4* src = (const float4*)(xW + (size_t)r * FEAT);
    float4 v = src[lane];                                      // 4 features per lane
    float* dst = out + (size_t)c * FEAT + (lane << 2);

    // Native L2 fp32 atomics (avoid any CAS-loop fallback).
    asm volatile(
        "global_atomic_add_f32 %0, %1, off\n\t"
        "global_atomic_add_f32 %0, %2, off offset:4\n\t"
        "global_atomic_add_f32 %0, %3, off offset:8\n\t"
        "global_atomic_add_f32 %0, %4, off offset:12"
        :
        : "v"(dst), "v"(nrm * v.x), "v"(nrm * v.y), "v"(nrm * v.z), "v"(nrm * v.w)
        : "memory");
}

extern "C" void kernel_launch(void* const* d_in, const int* in_sizes, int n_in,
                              void* d_out, int out_size, void* d_ws, size_t ws_size,
                              hipStream_t stream) {
    const float* x    = (const float*)d_in[0];
    const int*   ei   = (const int*)d_in[1];
    const float* W    = (const float*)d_in[2];
    const float* bias = (const float*)d_in[3];
    float*       out  = (float*)d_out;

    const int N = in_sizes[0] / FEAT;      // 100000
    const int E = in_sizes[1] / 2;         // 640000
    const int* row = ei;                   // edge_index[0] = sources
    const int* col = ei + E;               // edge_index[1] = targets

    // workspace layout: [cnt: N u32][dinv: N f32][xW: N*128 f32]  (~52 MB)
    unsigned* cnt  = (unsigned*)d_ws;
    float*    dinv = (float*)(cnt + N);
    float*    xW   = dinv + N;

    k_zero_u32 <<<(N + 255) / 256, 256, 0, stream>>>(cnt, N);
    k_count_deg<<<(E + 255) / 256, 256, 0, stream>>>(col, cnt, E);
    k_dinv     <<<(N + 255) / 256, 256, 0, stream>>>(cnt, dinv, N);
    k_gemm_xw  <<<(N + 15) / 16, 256, 0, stream>>>(x, W, xW, N);
    k_init_out <<<(N * FEAT + 255) / 256, 256, 0, stream>>>(xW, dinv, bias, out, N);
    k_aggregate<<<(E + 7) / 8, 256, 0, stream>>>(row, col, dinv, xW, out, E);
}